// CTCLoss_76759655514437
// MI455X (gfx1250) — compile-verified
//
#include <hip/hip_runtime.h>
#include <stdint.h>

#define Bdim 256
#define Tdim 512
#define Cdim 256
#define Ldim 64
#define Sdim 129            // 2L+1 extended states
#define ROWP 68             // padded lp row stride in floats: [blank, 64 labels, 3 pad] = 272B (17 x 16B)
#define LOG2E 1.4426950408889634f
#define LN2   0.6931471805599453f
#define NEG2  (-1.4426950e30f)   // -1e30 * log2(e): log-space 'zero' in base-2 domain

// ---------------------------------------------------------------------------
// Kernel 1: fused log2-softmax over C + gather of the 65 needed values.
// one wave per (b,t) row; 8 waves (256 threads) per block.
// Output lp'[j] = (logits[cls_j] - logsumexp) * log2(e)   (base-2 log-prob)
// ---------------------------------------------------------------------------
__global__ __launch_bounds__(256) void k1_lse_gather(
    const float* __restrict__ logits, const int* __restrict__ targets,
    float* __restrict__ lp)
{
  const int lane = threadIdx.x & 31;
  const int wv   = threadIdx.x >> 5;
  const int row  = blockIdx.x * 8 + wv;      // row in [0, B*T)
  const int b    = row >> 9;                 // row / T  (T = 512)

  const float* src = logits + (size_t)row * Cdim;

  __shared__ float rb[8][Cdim];

  float4 v0 = ((const float4*)src)[lane * 2 + 0];
  float4 v1 = ((const float4*)src)[lane * 2 + 1];

  float m = fmaxf(fmaxf(fmaxf(v0.x, v0.y), fmaxf(v0.z, v0.w)),
                  fmaxf(fmaxf(v1.x, v1.y), fmaxf(v1.z, v1.w)));
#pragma unroll
  for (int off = 16; off >= 1; off >>= 1)
    m = fmaxf(m, __shfl_xor(m, off, 32));

  // base-2 domain: s2 = sum 2^((x-m)*log2e);  L2 = log2(s2)
  float s = __builtin_amdgcn_exp2f((v0.x - m) * LOG2E)
          + __builtin_amdgcn_exp2f((v0.y - m) * LOG2E)
          + __builtin_amdgcn_exp2f((v0.z - m) * LOG2E)
          + __builtin_amdgcn_exp2f((v0.w - m) * LOG2E)
          + __builtin_amdgcn_exp2f((v1.x - m) * LOG2E)
          + __builtin_amdgcn_exp2f((v1.y - m) * LOG2E)
          + __builtin_amdgcn_exp2f((v1.z - m) * LOG2E)
          + __builtin_amdgcn_exp2f((v1.w - m) * LOG2E);
#pragma unroll
  for (int off = 16; off >= 1; off >>= 1)
    s += __shfl_xor(s, off, 32);

  const float L2 = __builtin_amdgcn_logf(s);   // v_log_f32 == log2

  ((float4*)&rb[wv][0])[lane * 2 + 0] = v0;
  ((float4*)&rb[wv][0])[lane * 2 + 1] = v1;
  __syncthreads();

  float* dst = lp + (size_t)row * ROWP;
  {
    int j   = lane;                                   // j = 0..31
    int cls = (j == 0) ? 0 : targets[b * Ldim + (j - 1)];
    dst[j]  = (rb[wv][cls] - m) * LOG2E - L2;
  }
  {
    int j   = lane + 32;                              // j = 32..63
    int cls = targets[b * Ldim + (j - 1)];
    dst[j]  = (rb[wv][cls] - m) * LOG2E - L2;
  }
  if (lane == 0) {
    int cls = targets[b * Ldim + 63];                 // j = 64
    dst[64] = (rb[wv][cls] - m) * LOG2E - L2;
  }
}

// ---------------------------------------------------------------------------
// CDNA5 async DMA of one 272B lp row into LDS (17 lanes x b128, ASYNCcnt)
// ---------------------------------------------------------------------------
__device__ __forceinline__ void async_row(const float* gsrc, const float* lbase,
                                          int lane)
{
  if (lane < 17) {
    // low 32 bits of a flat shared pointer == LDS byte offset (ISA 10.2 aperture)
    unsigned lds = (unsigned)(uintptr_t)lbase + (unsigned)(lane * 16);
    uint64_t ga  = (uint64_t)(uintptr_t)gsrc + (uint64_t)(lane * 16);
    asm volatile("global_load_async_to_lds_b128 %0, %1, off"
                 :: "v"(lds), "v"(ga)
                 : "memory");
  }
}

// ---------------------------------------------------------------------------
// Kernel 2: alpha recursion in base-2 log domain. One wave per batch row.
// Lane l owns states s = 32k + l (k = 0..4); neighbors via wave32 shuffles.
// lp rows prefetched 4 deep into an LDS ring via async DMA + s_wait_asynccnt.
// ---------------------------------------------------------------------------
__global__ __launch_bounds__(32) void k2_alpha(
    const float* __restrict__ lp, const int* __restrict__ targets,
    const int* __restrict__ in_len, const int* __restrict__ tgt_len,
    float* __restrict__ loss)
{
  const int b    = blockIdx.x;
  const int lane = threadIdx.x;
  const float* lpb = lp + (size_t)b * Tdim * ROWP;
  const int Tin = in_len[b];
  const int tl  = tgt_len[b];

  __shared__ float buf[4][ROWP];        // 4-deep ring, index = t & 3
  __shared__ float afin[5 * 32];

  // per-lane, per-k invariants
  int  jidx[5];
  bool act[5], skp[5];
#pragma unroll
  for (int k = 0; k < 5; ++k) {
    int  st  = k * 32 + lane;
    act[k]   = (st < Sdim);
    bool odd = (st & 1);
    int  i   = (st - 1) >> 1;
    jidx[k]  = odd ? (1 + i) : 0;          // index into lp row (0 = blank)
    bool sk  = false;
    if (act[k] && odd && st >= 3) {
      int la  = targets[b * Ldim + i];
      int lap = targets[b * Ldim + i - 1];
      sk = (la != lap);                    // labels are never blank in this setup
    }
    skp[k] = sk;
  }

  // t = 0 init: alpha[0]=lp'(blank), alpha[1]=lp'(label0), rest NEG2
  float alpha[5];
#pragma unroll
  for (int k = 0; k < 5; ++k) alpha[k] = NEG2;
  if (lane < 2) alpha[0] = lpb[lane];

  // prologue: fill the ring with rows 1..4
  async_row(lpb + 1 * ROWP, &buf[1][0], lane);
  async_row(lpb + 2 * ROWP, &buf[2][0], lane);
  async_row(lpb + 3 * ROWP, &buf[3][0], lane);
  async_row(lpb + 4 * ROWP, &buf[0][0], lane);

#pragma unroll 4
  for (int t = 1; t < Tdim; ++t) {
    // keep up to 3 rows in flight; guarantee row t is resident (in-order completion)
    asm volatile("s_wait_asynccnt 3" ::: "memory");
    const float* cur = &buf[t & 3][0];
    const bool upd = (t < Tin);

    float nalpha[5];
#pragma unroll
    for (int k = 0; k < 5; ++k) {
      float a1  = alpha[k];
      float up1 = __shfl_up(a1, 1, 32);
      float up2 = __shfl_up(a1, 2, 32);
      float c31 = (k > 0) ? __shfl(alpha[k - 1], 31, 32) : NEG2; // state 32k-1
      float a2  = (lane == 0) ? c31 : up1;                       // alpha[s-1]
      // skip is only ever true on odd lanes; lane1 needs state 32k-1 (= c31)
      float a3r = (lane >= 2) ? up2 : c31;
      float a3  = skp[k] ? a3r : NEG2;                           // alpha[s-2]
      float lpv = cur[jidx[k]];                                  // LDS gather
      float mm  = fmaxf(a1, fmaxf(a2, a3));
      // base-2 logaddexp3: all operands already scaled by log2(e)
      float r   = mm + __builtin_amdgcn_logf(__builtin_amdgcn_exp2f(a1 - mm)
                                           + __builtin_amdgcn_exp2f(a2 - mm)
                                           + __builtin_amdgcn_exp2f(a3 - mm)) + lpv;
      r = fmaxf(r, NEG2);                 // keep impossible paths bounded
      float na  = upd ? r : a1;           // freeze rows past input_length
      nalpha[k] = act[k] ? na : NEG2;
    }
#pragma unroll
    for (int k = 0; k < 5; ++k) alpha[k] = nalpha[k];

    if (t + 4 < Tdim)                     // refill the slot we just consumed
      async_row(lpb + (size_t)(t + 4) * ROWP, &buf[t & 3][0], lane);
  }

  // stage final alpha for the dynamic 2*tl index
#pragma unroll
  for (int k = 0; k < 5; ++k) afin[k * 32 + lane] = alpha[k];
  __syncthreads();

  if (lane == 0) {
    int   sl = 2 * tl;                    // <= 128
    float aL = afin[sl];
    float aP = afin[sl - 1];
    float mm = fmaxf(aL, aP);
    float l2 = mm + __builtin_amdgcn_logf(__builtin_amdgcn_exp2f(aL - mm)
                                        + __builtin_amdgcn_exp2f(aP - mm));
    float lo = -l2 * LN2;                 // back to natural-log domain
    if (lo > 1e20f) lo = 0.0f;            // zero_infinity
    loss[b] = lo / (float)tl;             // reduction='mean' per-sample scale
  }
}

// ---------------------------------------------------------------------------
// Kernel 3: deterministic batch mean -> scalar
// ---------------------------------------------------------------------------
__global__ __launch_bounds__(256) void k3_mean(const float* __restrict__ loss,
                                               float* __restrict__ out)
{
  __shared__ float red[256];
  int tid = threadIdx.x;
  red[tid] = loss[tid];
  __syncthreads();
#pragma unroll
  for (int off = 128; off >= 1; off >>= 1) {
    if (tid < off) red[tid] += red[tid + off];
    __syncthreads();
  }
  if (tid == 0) out[0] = red[0] * (1.0f / (float)Bdim);
}

// ---------------------------------------------------------------------------
extern "C" void kernel_launch(void* const* d_in, const int* in_sizes, int n_in,
                              void* d_out, int out_size, void* d_ws, size_t ws_size,
                              hipStream_t stream)
{
  const float* logits  = (const float*)d_in[0];   // [B,T,C] f32
  const int*   targets = (const int*)  d_in[1];   // [B,L]   i32
  const int*   in_len  = (const int*)  d_in[2];   // [B]     i32
  const int*   tgt_len = (const int*)  d_in[3];   // [B]     i32

  float* lpws    = (float*)d_ws;                              // B*T*ROWP floats (~35.7 MB)
  float* lossArr = lpws + (size_t)Bdim * Tdim * ROWP;         // + B floats
  float* out     = (float*)d_out;

  k1_lse_gather<<<(Bdim * Tdim) / 8, 256, 0, stream>>>(logits, targets, lpws);
  k2_alpha<<<Bdim, 32, 0, stream>>>(lpws, targets, in_len, tgt_len, lossArr);
  k3_mean<<<1, 256, 0, stream>>>(lossArr, out);
}